// GeoEncoder_71983651881269
// MI455X (gfx1250) — compile-verified
//
#include <hip/hip_runtime.h>

// -------- CDNA5 WMMA types --------
typedef __attribute__((ext_vector_type(2))) float v2f;
typedef __attribute__((ext_vector_type(8))) float v8f;

#define GRES 512
#define RANK 8
#define OUTC 8
#define F3R 24           // 3 * RANK
#define LDS_STRIDE 25    // 24 + 1 pad; gcd(25,64)=1 -> conflict-free strided reads
#define OUT_STRIDE 12    // 48B rows: 16B-aligned for b128, conflict-free writes

__device__ __forceinline__ float4 lerp4(float4 a, float4 b, float w) {
    float4 r;
    r.x = fmaf(w, b.x - a.x, a.x);
    r.y = fmaf(w, b.y - a.y, a.y);
    r.z = fmaf(w, b.z - a.z, a.z);
    r.w = fmaf(w, b.w - a.w, a.w);
    return r;
}

// Bilinear grid_sample, padding 'border', align_corners=False; writes 8 feats.
__device__ __forceinline__ void sample_plane(const float* __restrict__ plane,
                                             float u, float v,
                                             float* __restrict__ dst) {
    float gx = fminf(fmaxf(((u + 1.0f) * (float)GRES - 1.0f) * 0.5f, 0.0f),
                     (float)(GRES - 1));
    float gy = fminf(fmaxf(((v + 1.0f) * (float)GRES - 1.0f) * 0.5f, 0.0f),
                     (float)(GRES - 1));
    float x0f = floorf(gx), y0f = floorf(gy);
    float wx = gx - x0f, wy = gy - y0f;
    int x0 = (int)x0f, y0 = (int)y0f;
    int x1 = min(x0 + 1, GRES - 1);
    int y1 = min(y0 + 1, GRES - 1);

    const float4* r0 = (const float4*)(plane + (size_t)y0 * GRES * RANK);
    const float4* r1 = (const float4*)(plane + (size_t)y1 * GRES * RANK);
    // 8 x global_load_b128 per plane (planes L2-resident: 24MB in 192MB L2)
    float4 a0 = r0[x0 * 2], a1 = r0[x0 * 2 + 1];
    float4 b0 = r0[x1 * 2], b1 = r0[x1 * 2 + 1];
    float4 c0 = r1[x0 * 2], c1 = r1[x0 * 2 + 1];
    float4 d0 = r1[x1 * 2], d1 = r1[x1 * 2 + 1];

    float4 top0 = lerp4(a0, b0, wx), top1 = lerp4(a1, b1, wx);
    float4 bot0 = lerp4(c0, d0, wx), bot1 = lerp4(c1, d1, wx);
    float4 f0 = lerp4(top0, bot0, wy), f1 = lerp4(top1, bot1, wy);

    dst[0] = f0.x; dst[1] = f0.y; dst[2] = f0.z; dst[3] = f0.w;
    dst[4] = f1.x; dst[5] = f1.y; dst[6] = f1.z; dst[7] = f1.w;
}

__global__ __launch_bounds__(256)
void geo_encoder_kernel(const float* __restrict__ coords,
                        const float* __restrict__ pxy,
                        const float* __restrict__ pxz,
                        const float* __restrict__ pyz,
                        const float* __restrict__ W,     // [OUTC, F3R]
                        const float* __restrict__ bias,  // [OUTC]
                        float* __restrict__ out,         // [N, OUTC]
                        int N) {
    __shared__ float feats[8][32][LDS_STRIDE];    // 25.6 KB
    __shared__ float obuf[8][32][OUT_STRIDE];     // 12.3 KB

    const int tid  = threadIdx.x;
    const int wave = tid >> 5;
    const int lane = tid & 31;
    const int waveBase = blockIdx.x * 256 + wave * 32;

    const int n  = lane & 15;     // WMMA column (output channel for n<8)
    const int hi = lane >> 4;     // 0: lanes 0-15 (K base +0), 1: lanes 16-31 (+2)

    // --- hoisted, branch-free B-operand + bias loads (overlap the gathers) ---
    // clamped index keeps the load in-bounds & unconditional; zero via cndmask
    const int nc = min(n, OUTC - 1);
    const bool bvalid = (n < OUTC);
    v2f bv[6];
#pragma unroll
    for (int k = 0; k < 6; ++k) {
        const int kk = 4 * k + 2 * hi;                 // even -> 8B aligned
        float2 w = *(const float2*)(W + nc * F3R + kk); // global_load_b64
        bv[k].x = bvalid ? w.x : 0.0f;
        bv[k].y = bvalid ? w.y : 0.0f;
    }
    const float bn = bias[nc];

    // --- per-lane feature computation (index clamped for tail safety) ---
    int pc = min(waveBase + lane, N - 1);
    float cx = fminf(fmaxf(coords[(size_t)pc * 3 + 0], -1.0f), 1.0f);
    float cy = fminf(fmaxf(coords[(size_t)pc * 3 + 1], -1.0f), 1.0f);
    float cz = fminf(fmaxf(coords[(size_t)pc * 3 + 2], -1.0f), 1.0f);

    float* row = feats[wave][lane];
    sample_plane(pxy, cx, cy, row + 0);
    sample_plane(pxz, cx, cz, row + 8);
    sample_plane(pyz, cy, cz, row + 16);

    __syncthreads();

    // --- projection: two 16-point tiles per wave, 6 chained f32 WMMAs each ---
    // A layout (16x4 f32): lane L holds M=L%16; VGPR0 = K{0|2}, VGPR1 = K{1|3}
    v8f acc0 = {};
    v8f acc1 = {};
#pragma unroll
    for (int k = 0; k < 6; ++k) {
        const int kk = 4 * k + 2 * hi;
        v2f a0, a1;
        a0.x = feats[wave][n][kk];
        a0.y = feats[wave][n][kk + 1];
        a1.x = feats[wave][16 + n][kk];
        a1.y = feats[wave][16 + n][kk + 1];
        acc0 = __builtin_amdgcn_wmma_f32_16x16x4_f32(
            false, a0, false, bv[k], (short)0, acc0, false, false);
        acc1 = __builtin_amdgcn_wmma_f32_16x16x4_f32(
            false, a1, false, bv[k], (short)0, acc1, false, false);
    }

    // --- epilogue: bias + clamp, stage to LDS, coalesced b128 stores ---
    // D tile: VGPR r, lane L -> local point = tile*16 + r + 8*hi, chan = n
    if (bvalid) {
#pragma unroll
        for (int r = 0; r < 8; ++r) {
            obuf[wave][r + 8 * hi][n] =
                fminf(fmaxf(acc0[r] + bn, -10.0f), 10.0f);
            obuf[wave][16 + r + 8 * hi][n] =
                fminf(fmaxf(acc1[r] + bn, -10.0f), 10.0f);
        }
    }
    // same-wave LDS RAW: DS ops are in-order per wave; compiler inserts dscnt wait
    int p = waveBase + lane;
    if (p < N) {
        const float4* src = (const float4*)&obuf[wave][lane][0]; // 48B row, 16B aligned
        float4* dst = (float4*)(out + (size_t)p * OUTC);         // 32B aligned
        dst[0] = src[0];   // ds_load_b128 -> global_store_b128
        dst[1] = src[1];
    }
}

extern "C" void kernel_launch(void* const* d_in, const int* in_sizes, int n_in,
                              void* d_out, int out_size, void* d_ws, size_t ws_size,
                              hipStream_t stream) {
    const float* coords = (const float*)d_in[0];
    const float* pxy    = (const float*)d_in[1];
    const float* pxz    = (const float*)d_in[2];
    const float* pyz    = (const float*)d_in[3];
    const float* W      = (const float*)d_in[4];
    const float* b      = (const float*)d_in[5];
    float* out          = (float*)d_out;

    const int N = in_sizes[0] / 3;
    const int blocks = (N + 255) / 256;
    geo_encoder_kernel<<<blocks, 256, 0, stream>>>(coords, pxy, pxz, pyz, W, b,
                                                   out, N);
}